// MASNET_55851754717598
// MI455X (gfx1250) — compile-verified
//
#include <hip/hip_runtime.h>
#include <hip/hip_bf16.h>
#include <stdint.h>

// MASNET attention-sampled resize: B=32, C=3, H=W=512, out_size=512, dense=2.
// Bandwidth-bound gather (~0.03 FLOP/byte): no WMMA applies; the CDNA5 win is
// TDM tile staging (tensor_load_to_lds + s_wait_tensorcnt) + LDS gathers,
// B128 non-temporal streaming for the read-once/write-once traffic.

#ifndef __has_builtin
#define __has_builtin(x) 0
#endif

#if defined(__gfx1250__) && __has_builtin(__builtin_amdgcn_tensor_load_to_lds) && \
    __has_builtin(__builtin_amdgcn_s_wait_tensorcnt)
#define MAS_USE_TDM 1
#else
#define MAS_USE_TDM 0
#endif

typedef unsigned int v4u __attribute__((ext_vector_type(4)));
typedef int          v4i __attribute__((ext_vector_type(4)));
typedef int          v8i __attribute__((ext_vector_type(8)));
typedef float        v4f __attribute__((ext_vector_type(4)));

#define MAS_H 512
#define MAS_S 512
#define MAS_C 3

// ---------------------------------------------------------------------------
// Kernel A: marginals. grid (B, 8), 512 threads (16 waves). Each block owns 64
// rows of att[b]: per wave, a row is read as 4x v4f (global_load_b128,
// non-temporal: att is read exactly once). Row-max via wave32 shfl reduction;
// column-max via per-lane v4f accumulators -> LDS u32 atomicMax -> global
// u32 atomicMax (IEEE bit trick, valid since att >= 0 and marg_y pre-zeroed).
// Lane component j of chunk k covers column 128*k + 4*lane + j.
// ---------------------------------------------------------------------------
__global__ __launch_bounds__(512) void masnet_marginals(
    const float* __restrict__ att, float* __restrict__ marg_x,
    float* __restrict__ marg_y) {
  const int b = blockIdx.x;
  const int cy = blockIdx.y;  // 8 chunks of 64 rows
  const int tid = threadIdx.x;
  const int wv = tid >> 5, lane = tid & 31;

  __shared__ unsigned colmax_s[MAS_H];
  colmax_s[tid] = 0u;
  __syncthreads();

  v4f cp[4];
#pragma unroll
  for (int k = 0; k < 4; ++k) cp[k] = (v4f)(0.0f);

#pragma unroll
  for (int g = 0; g < 4; ++g) {
    const int h = cy * 64 + g * 16 + wv;
    const v4f* row4 = (const v4f*)(att + ((size_t)b * MAS_H + h) * MAS_H);
    if (g < 3) __builtin_prefetch(row4 + 4 * MAS_H + lane, 0, 0);  // next group
    float m = 0.0f;
#pragma unroll
    for (int k = 0; k < 4; ++k) {
      v4f v = __builtin_nontemporal_load(row4 + lane + 32 * k);
      cp[k].x = fmaxf(cp[k].x, v.x);
      cp[k].y = fmaxf(cp[k].y, v.y);
      cp[k].z = fmaxf(cp[k].z, v.z);
      cp[k].w = fmaxf(cp[k].w, v.w);
      m = fmaxf(m, fmaxf(fmaxf(v.x, v.y), fmaxf(v.z, v.w)));
    }
#pragma unroll
    for (int off = 16; off >= 1; off >>= 1) m = fmaxf(m, __shfl_xor(m, off, 32));
    if (lane == 0) marg_x[b * MAS_H + h] = m;  // row uniquely owned: plain store
  }
#pragma unroll
  for (int k = 0; k < 4; ++k) {
    const int col = 128 * k + 4 * lane;
    atomicMax(&colmax_s[col + 0], __float_as_uint(cp[k].x));
    atomicMax(&colmax_s[col + 1], __float_as_uint(cp[k].y));
    atomicMax(&colmax_s[col + 2], __float_as_uint(cp[k].z));
    atomicMax(&colmax_s[col + 3], __float_as_uint(cp[k].w));
  }
  __syncthreads();
  atomicMax((unsigned*)(marg_y) + b * MAS_H + tid, colmax_s[tid]);
}

// ---------------------------------------------------------------------------
// Kernel B: inverse-CDF positions. grid (B, 2), 512 threads. axis 0: marg_x ->
// pos_x ; axis 1: marg_y -> pos_y. Block-sum reductions, Hillis-Steele scan in
// LDS, per-thread binary search (searchsorted 'left'), bin interpolation.
// ---------------------------------------------------------------------------
__device__ __forceinline__ float block_sum512(float v, volatile float* red) {
#pragma unroll
  for (int off = 16; off >= 1; off >>= 1) v += __shfl_xor(v, off, 32);
  const int wv = threadIdx.x >> 5, lane = threadIdx.x & 31;
  __syncthreads();
  if (lane == 0) red[wv] = v;
  __syncthreads();
  if (threadIdx.x == 0) {
    float s = 0.0f;
#pragma unroll
    for (int i = 0; i < 16; ++i) s += red[i];
    red[0] = s;
  }
  __syncthreads();
  return red[0];
}

__global__ __launch_bounds__(512) void masnet_inverse_cdf(
    const float* __restrict__ marg_x, const float* __restrict__ marg_y,
    float* __restrict__ pos_x, float* __restrict__ pos_y,
    const int* __restrict__ dense_p) {
  const int b = blockIdx.x;
  const int axis = blockIdx.y;
  const int tid = threadIdx.x;
  __shared__ float c_lds[MAS_H];
  __shared__ float red[16];

  const float dense = (float)dense_p[0];
  const float* m = (axis == 0) ? marg_x : marg_y;
  float* pout = (axis == 0) ? pos_x : pos_y;

  float a = m[b * MAS_H + tid];
  float s = block_sum512(a, red);
  a = a * ((float)MAS_S / s);
#pragma unroll
  for (int it = 0; it < 5; ++it) {  // ITERS = 5
    a = fminf(a, dense);
    s = block_sum512(a, red);
    a = a * ((float)MAS_S / s);
  }

  c_lds[tid] = a;
  __syncthreads();
  for (int off = 1; off < MAS_H; off <<= 1) {  // inclusive scan
    float add = (tid >= off) ? c_lds[tid - off] : 0.0f;
    __syncthreads();
    c_lds[tid] += add;
    __syncthreads();
  }

  const float t = (float)tid + 0.5f;
  int lo = 0, hi = MAS_H;  // lower_bound: first idx with c[idx] >= t
  while (lo < hi) {
    int mid = (lo + hi) >> 1;
    if (c_lds[mid] < t) lo = mid + 1; else hi = mid;
  }
  int idx = (lo < MAS_H - 1) ? lo : (MAS_H - 1);
  float cc = c_lds[idx];
  float cprev = (idx > 0) ? c_lds[idx - 1] : 0.0f;
  float frac = (t - cprev) / fmaxf(cc - cprev, 1e-6f);
  float pos = (float)idx - 0.5f + frac;
  pos = fminf(fmaxf(pos, 0.0f), (float)(MAS_H - 1));
  pout[b * MAS_H + tid] = pos;
}

// ---------------------------------------------------------------------------
// Kernel C: separable bilinear gather. grid (512 rows, B), 512 threads.
// One TDM op stages the 512x2x3 source tile (12KB) into LDS; lanes gather
// 4 taps/channel from LDS (monotone p0x => near conflict-free) and write
// coalesced 2KB output rows with non-temporal stores (output never re-read,
// keeps the 192MB L2 free for the data tiles the TDM re-reads).
// ---------------------------------------------------------------------------
__global__ __launch_bounds__(512) void masnet_resample(
    const float* __restrict__ data, const float* __restrict__ pos_x,
    const float* __restrict__ pos_y, float* __restrict__ out) {
  __shared__ float rows[MAS_C * 2 * MAS_H];  // [c][r][x], contiguous: TDM order
  const int i = blockIdx.x;
  const int b = blockIdx.y;
  const int tid = threadIdx.x;

  const float y = pos_y[b * MAS_S + i];
  int p0y = (int)floorf(y);
  p0y = p0y < 0 ? 0 : (p0y > MAS_H - 2 ? MAS_H - 2 : p0y);
  const float wy = y - (float)p0y;
  const float* src =
      data + (size_t)b * MAS_C * MAS_H * MAS_H + (size_t)p0y * MAS_H;

#if MAS_USE_TDM
  if (tid < 32) {  // one wave issues one 3D TDM: X=512 cols, Y=2 rows, Z=3 ch
    unsigned long long ga = (unsigned long long)(size_t)src;
    v4u g0;
    g0[0] = 1u;                         // count=1, user descriptor
    g0[1] = 0u;                         // lds_addr: 'rows' is at LDS offset 0
    g0[2] = (unsigned)ga;               // global_addr[31:0]
    g0[3] = ((unsigned)(ga >> 32) & 0x01FFFFFFu) | (2u << 30);  // addr[56:32], type=2
    v8i g1;
    g1[0] = (int)(2u << 16);            // data_size = 4B (code 2)
    g1[1] = (int)(512u << 16);          // tensor_dim0[15:0] = 512
    g1[2] = (int)(2u << 16);            // tensor_dim1[15:0] = 2 (rows valid)
    g1[3] = (int)(512u << 16);          // tile_dim0 = 512
    g1[4] = (int)(2u | (3u << 16));     // tile_dim1 = 2, tile_dim2 = 3
    g1[5] = 512;                        // tensor_dim0_stride = 512 elems (row)
    g1[6] = 0;                          // dim0_stride hi, dim1_stride[15:0]=0
    g1[7] = 4;                          // dim1_stride[47:16]: 262144>>16 (chan)
    v4i g2; g2[0] = 3; g2[1] = 0; g2[2] = 0; g2[3] = 0;  // tensor_dim2 = 3
    v4i g3; g3[0] = 0; g3[1] = 0; g3[2] = 0; g3[3] = 0;
    v8i g4;                              // VADDR4 slot: unused per ISA -> zeros
    g4[0] = 0; g4[1] = 0; g4[2] = 0; g4[3] = 0;
    g4[4] = 0; g4[5] = 0; g4[6] = 0; g4[7] = 0;
    __builtin_amdgcn_tensor_load_to_lds(g0, g1, g2, g3, g4, 0);
    __builtin_amdgcn_s_wait_tensorcnt(0);
  }
  __syncthreads();
#else
#pragma unroll
  for (int c = 0; c < MAS_C; ++c) {
    rows[c * 1024 + tid] = src[(size_t)c * MAS_H * MAS_H + tid];
    rows[c * 1024 + MAS_H + tid] = src[(size_t)c * MAS_H * MAS_H + MAS_H + tid];
  }
  __syncthreads();
#endif

  const float x = pos_x[b * MAS_S + tid];
  int p0x = (int)floorf(x);
  p0x = p0x < 0 ? 0 : (p0x > MAS_H - 2 ? MAS_H - 2 : p0x);
  const float wx = x - (float)p0x;
  const float w00 = (1.0f - wy) * (1.0f - wx), w01 = (1.0f - wy) * wx;
  const float w10 = wy * (1.0f - wx), w11 = wy * wx;
#pragma unroll
  for (int c = 0; c < MAS_C; ++c) {
    const float* r0 = &rows[c * 1024];
    float v = w00 * r0[p0x] + w01 * r0[p0x + 1] +
              w10 * r0[MAS_H + p0x] + w11 * r0[MAS_H + p0x + 1];
    __builtin_nontemporal_store(
        v, out + (((size_t)b * MAS_C + c) * MAS_S + i) * MAS_S + tid);
  }
}

// ---------------------------------------------------------------------------
extern "C" void kernel_launch(void* const* d_in, const int* in_sizes, int n_in,
                              void* d_out, int out_size, void* d_ws,
                              size_t ws_size, hipStream_t stream) {
  const float* data = (const float*)d_in[0];
  const float* att = (const float*)d_in[1];
  const int* dense_p = (const int*)d_in[3];  // out_size is d_in[2] (fixed 512)
  float* out = (float*)d_out;

  const int B = in_sizes[1] / (MAS_H * MAS_H);  // att is [B, 512, 512]

  float* marg_x = (float*)d_ws;                 // [B,512] row maxes  -> pos_x
  float* marg_y = marg_x + (size_t)B * MAS_H;   // [B,512] col maxes  -> pos_y
  float* pos_x = marg_y + (size_t)B * MAS_H;    // [B,512]
  float* pos_y = pos_x + (size_t)B * MAS_H;     // [B,512]

  (void)hipMemsetAsync(marg_y, 0, (size_t)B * MAS_H * sizeof(float), stream);
  masnet_marginals<<<dim3(B, 8), 512, 0, stream>>>(att, marg_x, marg_y);
  masnet_inverse_cdf<<<dim3(B, 2), 512, 0, stream>>>(marg_x, marg_y, pos_x,
                                                     pos_y, dense_p);
  masnet_resample<<<dim3(MAS_S, B), 512, 0, stream>>>(data, pos_x, pos_y, out);
}